// TwoStepSpodNet_61692910239923
// MI455X (gfx1250) — compile-verified
//
#include <hip/hip_runtime.h>

typedef __attribute__((ext_vector_type(2))) float v2f;
typedef __attribute__((ext_vector_type(8))) float v8f;

#define P   64
#define PS  65      // padded LDS row stride (bank-conflict-free column access)
#define NT  256     // 8 wave32s per workgroup

__global__ __launch_bounds__(NT)
void spodnet_kernel(const float* __restrict__ ThetaG,
                    const float* __restrict__ W1g,
                    const float* __restrict__ b1g,
                    const float* __restrict__ W2g,
                    const float* __restrict__ b2g,
                    const float* __restrict__ A1g,
                    const float* __restrict__ ab1g,
                    const float* __restrict__ A2g,
                    const float* __restrict__ ab2g,
                    const float* __restrict__ A3g,
                    const float* __restrict__ ab3g,
                    float* __restrict__ OutG)
{
    __shared__ float Th[P*PS];      // Theta (updated in place)
    __shared__ float Am[P*PS];      // Cholesky L, then W = inv(Theta)
    __shared__ float Mi[P*PS];      // L^{-1}
    __shared__ float ph[64*10];     // partial sums for h init
    __shared__ float W2s[10*63];
    __shared__ float b2s[63];
    __shared__ float b1s[10];
    __shared__ float hacc[10];      // running flat(Th) @ W1
    __shared__ float hr[10];        // relu(hacc + b1)
    __shared__ float dl[63];        // per-column delta
    __shared__ float uv[P], yv[P], pt[P], zv[P], ow[P];
    __shared__ float alw[32];       // A1(6) ab1(3) A2(9) ab2(3) A3(3) ab3(1)
    __shared__ float sc[8];         // scalar broadcast: s1, w22, gy, 1/gy

    const int tid = threadIdx.x;
    const int b   = blockIdx.x;
    const float* Tb = ThetaG + (size_t)b * P * P;

    __builtin_prefetch(Tb, 0, 1);   // global_prefetch_b8

    // ---------- load Theta -> LDS via 128-bit global loads ----------
    // 1024 float4 chunks; row = q>>4, col0 = (q&15)*4. Global side is b128
    // (HBM-bound path); LDS side scalar (row stride 65 is not 16B aligned,
    // compiler pairs them into ds_store_2addr).
    {
        const float4* Tb4 = (const float4*)Tb;
        for (int q = tid; q < (P*P)/4; q += NT) {
            float4 v = Tb4[q];
            int base = (q >> 4)*PS + ((q & 15) << 2);
            Th[base+0] = v.x; Th[base+1] = v.y; Th[base+2] = v.z; Th[base+3] = v.w;
            Am[base+0] = v.x; Am[base+1] = v.y; Am[base+2] = v.z; Am[base+3] = v.w;
            Mi[base+0] = 0.f; Mi[base+1] = 0.f; Mi[base+2] = 0.f; Mi[base+3] = 0.f;
        }
    }
    for (int i = tid; i < 630; i += NT) W2s[i] = W2g[i];
    if (tid < 63) b2s[tid]    = b2g[tid];
    if (tid < 10) b1s[tid]    = b1g[tid];
    if (tid < 6)  alw[tid]    = A1g[tid];
    if (tid < 3)  alw[6+tid]  = ab1g[tid];
    if (tid < 9)  alw[9+tid]  = A2g[tid];
    if (tid < 3)  alw[18+tid] = ab2g[tid];
    if (tid < 3)  alw[21+tid] = A3g[tid];
    if (tid == 0) alw[24]     = ab3g[0];
    __syncthreads();

    // ---------- h_init: hacc[m] = sum_idx Th_flat[idx] * W1[idx][m] ----------
    if (tid < 64) {
        float acc[10];
        #pragma unroll
        for (int m = 0; m < 10; ++m) acc[m] = 0.f;
        for (int idx = tid; idx < P*P; idx += 64) {
            float v = Th[(idx>>6)*PS + (idx&63)];
            const float* wrow = W1g + (size_t)idx * 10;
            #pragma unroll
            for (int m = 0; m < 10; ++m) acc[m] += v * wrow[m];
        }
        #pragma unroll
        for (int m = 0; m < 10; ++m) ph[tid*10 + m] = acc[m];
    }
    __syncthreads();
    if (tid < 10) {
        float a = 0.f;
        for (int t = 0; t < 64; ++t) a += ph[t*10 + tid];
        hacc[tid] = a;
    }
    __syncthreads();

    // ---------- Cholesky: Am(lower) = L with Theta = L L^T ----------
    for (int k = 0; k < P; ++k) {
        if (tid == 0) Am[k*PS + k] = sqrtf(Am[k*PS + k]);
        __syncthreads();
        float d = Am[k*PS + k];
        for (int i = k + 1 + tid; i < P; i += NT) Am[i*PS + k] /= d;
        __syncthreads();
        for (int e = tid; e < P*P; e += NT) {
            int i = e >> 6, j = e & 63;
            if (i > k && j > k && j <= i)
                Am[i*PS + j] -= Am[i*PS + k] * Am[j*PS + k];
        }
        __syncthreads();
    }

    // ---------- Mi = L^{-1} (64 independent columns, forward substitution) ----------
    if (tid < P) {
        const int c = tid;
        Mi[c*PS + c] = 1.0f / Am[c*PS + c];
        for (int i = c + 1; i < P; ++i) {
            float s = 0.f;
            for (int j = c; j < i; ++j) s += Am[i*PS + j] * Mi[j*PS + c];
            Mi[i*PS + c] = -s / Am[i*PS + i];
        }
    }
    __syncthreads();

    // ---------- W = Mi^T * Mi  via V_WMMA_F32_16X16X4_F32, written into Am ----------
    // D[m][n] = sum_k Mi[k][I+m] * Mi[k][J+n]; 16 tiles over 8 waves.
    // Tile pair (w, w+8) shares the same J block -> shared B fragments.
    {
        const int wid  = tid >> 5;
        const int lane = tid & 31;
        const int half = lane >> 4;    // lane 0-15 vs 16-31
        const int mn   = lane & 15;
        const int I0 = ((wid >> 2) + 0) * 16;   // tile wid
        const int I1 = ((wid >> 2) + 2) * 16;   // tile wid+8 (same J)
        const int J  = (wid & 3) * 16;
        v8f c0 = {0.f,0.f,0.f,0.f,0.f,0.f,0.f,0.f};
        v8f c1 = {0.f,0.f,0.f,0.f,0.f,0.f,0.f,0.f};
        #pragma unroll
        for (int k0 = 0; k0 < P; k0 += 4) {
            // A 16x4 layout: lanes 0-15 hold K={0,1}; lanes 16-31 hold K={2,3}
            const int r0 = (k0 + 2*half + 0)*PS;
            const int r1 = (k0 + 2*half + 1)*PS;
            v2f a0, a1, bb;
            a0[0] = Mi[r0 + I0 + mn];  a0[1] = Mi[r1 + I0 + mn];
            a1[0] = Mi[r0 + I1 + mn];  a1[1] = Mi[r1 + I1 + mn];
            // B 4x16 row-striped: VGPR0 = K{0|2}, VGPR1 = K{1|3}
            bb[0] = Mi[r0 + J + mn];   bb[1] = Mi[r1 + J + mn];
            c0 = __builtin_amdgcn_wmma_f32_16x16x4_f32(
                     false, a0, false, bb, (short)0, c0, false, false);
            c1 = __builtin_amdgcn_wmma_f32_16x16x4_f32(
                     false, a1, false, bb, (short)0, c1, false, false);
        }
        // C/D layout: VGPR r -> M = r (lanes 0-15) or r+8 (lanes 16-31)
        #pragma unroll
        for (int r = 0; r < 8; ++r) {
            Am[(I0 + r + 8*half)*PS + J + mn] = c0[r];
            Am[(I1 + r + 8*half)*PS + J + mn] = c1[r];
        }
    }
    __syncthreads();

    // ---------- Pass 1: 64 sequential column sweeps ----------
    for (int col = 0; col < P; ++col) {
        if (tid < 10) hr[tid] = fmaxf(hacc[tid] + b1s[tid], 0.f);
        __syncthreads();
        if (tid < 63) {
            const int j   = tid;
            const int mcj = j + (j >= col);
            float o = b2s[j];
            #pragma unroll
            for (int m = 0; m < 10; ++m) o += hr[m] * W2s[m*63 + j];
            float old = Th[mcj*PS + col];
            dl[j] = o - old;
            Th[mcj*PS + col] = o;     // symmetric column/row rewrite
            Th[col*PS + mcj] = o;
        }
        __syncthreads();
        if (tid < 10) {               // incremental h update (L2-resident W1 rows)
            const int m = tid;
            float a = 0.f;
            for (int j = 0; j < 63; ++j) {
                const int mcj = j + (j >= col);
                a += dl[j] * (W1g[(size_t)(mcj*64 + col)*10 + m] +
                              W1g[(size_t)(col*64 + mcj)*10 + m]);
            }
            hacc[m] += a;
        }
        __syncthreads();
    }

    // ---------- Pass 2: 64 sequential column sweeps on (Th, W=Am) ----------
    for (int col = 0; col < P; ++col) {
        if (tid < P) uv[tid] = (tid == col) ? 0.f : Th[tid*PS + col];
        __syncthreads();
        if (tid < P) {
            float y = 0.f;
            for (int j = 0; j < P; ++j) y += Am[tid*PS + j] * uv[j];
            yv[tid] = y;
            pt[tid] = y * uv[tid];
            ow[tid] = Am[tid*PS + col];   // old w12 (and w22 at tid==col)
        }
        __syncthreads();
        if (tid == 0) {
            float s1  = yv[col];                 // w12 . theta12
            float w22 = Am[col*PS + col];
            float sty = 0.f;
            for (int i = 0; i < P; ++i) sty += pt[i];
            float schur = sty - s1*s1 / w22;     // theta12^T inv11 theta12
            float t22 = Th[col*PS + col];
            // alpha MLP: (w22,t22) -> 3 -> 3 -> 1, exp
            float h1[3], h2[3];
            #pragma unroll
            for (int a = 0; a < 3; ++a)
                h1[a] = fmaxf(w22*alw[a] + t22*alw[3+a] + alw[6+a], 0.f);
            #pragma unroll
            for (int bi = 0; bi < 3; ++bi) {
                float s = alw[18+bi];
                #pragma unroll
                for (int a = 0; a < 3; ++a) s += h1[a]*alw[9 + a*3 + bi];
                h2[bi] = fmaxf(s, 0.f);
            }
            float g = alw[24];
            #pragma unroll
            for (int a = 0; a < 3; ++a) g += h2[a]*alw[21+a];
            float gy = expf(g);
            Th[col*PS + col] = gy + schur;
            sc[0] = s1; sc[1] = w22; sc[2] = gy; sc[3] = 1.0f/gy;
        }
        __syncthreads();
        if (tid < P) {
            if (tid == col) zv[tid] = 0.f;
            else {
                float z = yv[tid] - (sc[0]/sc[1]) * ow[tid]; // inv11 @ theta12
                zv[tid] = -sc[3] * z;                         // w12_next
            }
        }
        __syncthreads();
        {
            const float w22 = sc[1], gy = sc[2], w22n = sc[3];
            const float iw22 = 1.0f / w22;
            for (int e = tid; e < P*P; e += NT) {
                int i = e >> 6, j = e & 63;
                if (i == col && j == col)      Am[i*PS + j]  = w22n;
                else if (i == col)             Am[i*PS + j]  = zv[j];
                else if (j == col)             Am[i*PS + j]  = zv[i];
                else                           Am[i*PS + j] += -ow[i]*ow[j]*iw22
                                                               + gy*zv[i]*zv[j];
            }
        }
        __syncthreads();
    }

    // ---------- write Theta back via 128-bit global stores ----------
    {
        float4* Ob4 = (float4*)(OutG + (size_t)b * P * P);
        for (int q = tid; q < (P*P)/4; q += NT) {
            int base = (q >> 4)*PS + ((q & 15) << 2);
            float4 v;
            v.x = Th[base+0]; v.y = Th[base+1]; v.z = Th[base+2]; v.w = Th[base+3];
            Ob4[q] = v;
        }
    }
}

extern "C" void kernel_launch(void* const* d_in, const int* in_sizes, int n_in,
                              void* d_out, int out_size, void* d_ws, size_t ws_size,
                              hipStream_t stream) {
    const float* Theta = (const float*)d_in[0];
    const float* W1    = (const float*)d_in[1];
    const float* b1    = (const float*)d_in[2];
    const float* W2    = (const float*)d_in[3];
    const float* b2    = (const float*)d_in[4];
    const float* A1    = (const float*)d_in[5];
    const float* ab1   = (const float*)d_in[6];
    const float* A2    = (const float*)d_in[7];
    const float* ab2   = (const float*)d_in[8];
    const float* A3    = (const float*)d_in[9];
    const float* ab3   = (const float*)d_in[10];
    float* Out = (float*)d_out;

    const int B = in_sizes[0] / (P * P);   // 4096
    spodnet_kernel<<<dim3(B), dim3(NT), 0, stream>>>(
        Theta, W1, b1, W2, b2, A1, ab1, A2, ab2, A3, ab3, Out);
}